// ResidualBlock5_43568148250738
// MI455X (gfx1250) — compile-verified
//
#include <hip/hip_runtime.h>

typedef __attribute__((ext_vector_type(2))) float v2f;
typedef __attribute__((ext_vector_type(8))) float v8f;

#define Lpos 8192
#define Cch  32
#define Fch  32
#define BN_EPS 1e-3f
#define ALPHA  0.3f

__device__ __forceinline__ float leaky(float v) { return v >= 0.0f ? v : ALPHA * v; }

// One wave32 computes the full [32 x 32] output tile for one position l:
//   out[b,l,f] = 2*bias[l,f] + sum over taps of x[tap] @ W[tap,l]
// then fused BN + LeakyReLU (+ optional residual).
// All indices are unsigned 32-bit so the backend can emit saddr-form global
// loads (SGPR base + 32-bit voffset) and fold K-step deltas into ioffset.
__launch_bounds__(256)
__global__ void lc_bn_leaky_kernel(const float* __restrict__ in,    // [B,L,C]
                                   const float* __restrict__ w,     // [K,L,C,F]
                                   const float* __restrict__ bias,  // [L,F]
                                   const float* __restrict__ gamma, // [F]
                                   const float* __restrict__ beta,  // [F]
                                   const float* __restrict__ mean,  // [F]
                                   const float* __restrict__ var,   // [F]
                                   const float* __restrict__ resid, // [B,L,F] or nullptr
                                   float* __restrict__ out)         // [B,L,F]
{
    const int lane  = threadIdx.x & 31;
    const int wave  = threadIdx.x >> 5;
    const int pos   = blockIdx.x * (blockDim.x >> 5) + wave;
    const int lmod  = lane & 15;        // M (for A) / N (for B,C,D)
    const int hi    = lane >> 4;        // 0: lanes 0-15, 1: lanes 16-31
    const int khalf = hi << 1;          // K offset within a K=4 step

    // Accumulators: acc[mtile][ntile], each 16x16 f32 in 8 VGPRs.
    v8f acc[2][2];
    #pragma unroll
    for (int n = 0; n < 2; ++n) {
        // Reference adds bias twice.
        float bv = 2.0f * bias[(unsigned)(pos * Fch + 16 * n + lmod)];
        #pragma unroll
        for (int r = 0; r < 8; ++r) { acc[0][n][r] = bv; acc[1][n][r] = bv; }
    }

    // Position-offset groups; kernels 0,1 both read x[l] (roll step 0 twice),
    // kernel 2 -> x[l-1], kernel 3 -> x[l+1], kernel 4 -> x[l-2].
    const int offs[4]   = { 0, -1, 1, -2 };
    const int wfirst[4] = { 0,  2, 3, 4 };
    const int wcnt[4]   = { 2,  1, 1, 1 };

    #pragma unroll
    for (int g = 0; g < 4; ++g) {
        const unsigned ptap = (unsigned)((pos + offs[g] + Lpos) & (Lpos - 1));
        #pragma unroll
        for (int ks = 0; ks < 8; ++ks) {
            const unsigned c = 4u * (unsigned)ks + (unsigned)khalf;  // lane K indices: c, c+1

            // A tiles (shared by all weight blocks of this group):
            // A[M=batch, K=channel]; VGPR0=K(c), VGPR1=K(c+1); lane%16 = M.
            v2f a[2];
            #pragma unroll
            for (int m = 0; m < 2; ++m) {
                const unsigned ai =
                    (unsigned)(16 * m + lmod) * (unsigned)(Lpos * Cch) + ptap * Cch + c;
                a[m][0] = in[ai];
                a[m][1] = in[ai + 1u];
            }

            #pragma unroll
            for (int j = 0; j < 2; ++j) {
                if (j < wcnt[g]) {   // constant-folds under full unroll
                    const unsigned wb =
                        ((unsigned)(wfirst[g] + j) * (unsigned)Lpos + (unsigned)pos)
                        * (unsigned)(Cch * Fch);
                    // B tiles: B[K=channel, N=filter]; VGPR0=K(c), VGPR1=K(c+1); lane%16 = N.
                    v2f bm[2];
                    #pragma unroll
                    for (int n = 0; n < 2; ++n) {
                        const unsigned wi = wb + c * Fch + (unsigned)(16 * n + lmod);
                        bm[n][0] = w[wi];
                        bm[n][1] = w[wi + (unsigned)Fch];
                    }
                    #pragma unroll
                    for (int m = 0; m < 2; ++m)
                        #pragma unroll
                        for (int n = 0; n < 2; ++n)
                            acc[m][n] = __builtin_amdgcn_wmma_f32_16x16x4_f32(
                                false, a[m], false, bm[n], (short)0, acc[m][n], false, false);
                }
            }
        }
    }

    // Epilogue: BN + LeakyReLU (+ residual), write [B,L,F].
    #pragma unroll
    for (int n = 0; n < 2; ++n) {
        const int f = 16 * n + lmod;
        const float sc = gamma[f] * rsqrtf(var[f] + BN_EPS);
        const float sh = beta[f] - mean[f] * sc;
        #pragma unroll
        for (int m = 0; m < 2; ++m) {
            #pragma unroll
            for (int r = 0; r < 8; ++r) {
                const unsigned b = (unsigned)(16 * m + r + 8 * hi);  // C/D: VGPR r -> M=r (+8 hi)
                const unsigned oi = (b * (unsigned)Lpos + (unsigned)pos) * (unsigned)Fch
                                    + (unsigned)f;
                float v = acc[m][n][r] * sc + sh;
                if (resid) v = leaky(v + resid[oi]);
                else       v = leaky(v);
                out[oi] = v;
            }
        }
    }
}

extern "C" void kernel_launch(void* const* d_in, const int* in_sizes, int n_in,
                              void* d_out, int out_size, void* d_ws, size_t ws_size,
                              hipStream_t stream) {
    const float* x   = (const float*)d_in[0];
    const float* k1  = (const float*)d_in[1];
    const float* b1  = (const float*)d_in[2];
    const float* g1  = (const float*)d_in[3];
    const float* be1 = (const float*)d_in[4];
    const float* m1  = (const float*)d_in[5];
    const float* v1  = (const float*)d_in[6];
    const float* k2  = (const float*)d_in[7];
    const float* b2  = (const float*)d_in[8];
    const float* g2  = (const float*)d_in[9];
    const float* be2 = (const float*)d_in[10];
    const float* m2  = (const float*)d_in[11];
    const float* v2  = (const float*)d_in[12];

    float* h   = (float*)d_ws;   // [B,L,F] intermediate: 32*8192*32*4 = 32 MiB
    float* out = (float*)d_out;

    dim3 block(256);             // 8 waves -> 8 positions per block
    dim3 grid(Lpos / 8);         // 1024 blocks

    // Layer 1: h = leaky(bn1(lc1(x)))
    lc_bn_leaky_kernel<<<grid, block, 0, stream>>>(x, k1, b1, g1, be1, m1, v1, nullptr, h);
    // Layer 2: out = leaky(bn2(lc2(h)) + x)
    lc_bn_leaky_kernel<<<grid, block, 0, stream>>>(h, k2, b2, g2, be2, m2, v2, x, out);
}